// NarrativeGraphLayer_63101659513093
// MI455X (gfx1250) — compile-verified
//
#include <hip/hip_runtime.h>
#include <hip/hip_bf16.h>

typedef __attribute__((ext_vector_type(2))) float v2f;
typedef __attribute__((ext_vector_type(8))) float v8f;

#define CIN  128
#define COUT 128
#define LDS_STRIDE 132   // pad 128->132 floats: conflict-free for strided A reads

// ---------------------------------------------------------------------------
// Kernel 1: init accumulator to 0 and degree to 1 (self-loop included)
// ---------------------------------------------------------------------------
__global__ void gcn_init(float* __restrict__ agg, unsigned int* __restrict__ deg,
                         int N) {
    long long i = (long long)blockIdx.x * blockDim.x + threadIdx.x;
    long long total = (long long)N * CIN;
    if (i < total) agg[i] = 0.0f;
    if (i < N) deg[i] = 1u;   // self-loop contributes 1 to degree
}

// ---------------------------------------------------------------------------
// Kernel 2: integer degree count on dst (exact, deterministic)
// ---------------------------------------------------------------------------
__global__ void gcn_degree(const int* __restrict__ dst, unsigned int* __restrict__ deg,
                           int E) {
    int e = blockIdx.x * blockDim.x + threadIdx.x;
    if (e < E) atomicAdd(&deg[dst[e]], 1u);
}

// ---------------------------------------------------------------------------
// Kernel 3: deg -> deg^{-1/2}, in place (uint read, float write, same slot)
// ---------------------------------------------------------------------------
__global__ void gcn_rsqrt(float* __restrict__ degdis, int N) {
    int i = blockIdx.x * blockDim.x + threadIdx.x;
    if (i < N) {
        unsigned int d = ((const unsigned int*)degdis)[i];
        degdis[i] = rsqrtf((float)d);   // d >= 1 always (self-loop)
    }
}

// ---------------------------------------------------------------------------
// Kernel 4: edge scatter. One wave32 per edge: float4 gather of the 512B
// source row (coalesced), scale by norm, 4 fp32 atomics per lane into the
// L2-resident accumulator. All hot data (x: 51MB, agg: 51MB) fits in 192MB L2.
// ---------------------------------------------------------------------------
__global__ __launch_bounds__(256)
void gcn_scatter(const float* __restrict__ x,
                 const int* __restrict__ src, const int* __restrict__ dst,
                 const float* __restrict__ dis, float* __restrict__ agg, int E) {
    long long gid  = (long long)blockIdx.x * blockDim.x + threadIdx.x;
    long long edge = gid >> 5;
    int lane = (int)(gid & 31);
    if (edge >= E) return;
    int s = src[edge];
    int d = dst[edge];
    float norm = dis[s] * dis[d];
    const float4* xr = (const float4*)(x + (size_t)s * CIN);
    float4 v = xr[lane];
    float* ar = agg + (size_t)d * CIN + lane * 4;
    atomicAdd(ar + 0, v.x * norm);
    atomicAdd(ar + 1, v.y * norm);
    atomicAdd(ar + 2, v.z * norm);
    atomicAdd(ar + 3, v.w * norm);
}

// ---------------------------------------------------------------------------
// Kernel 5: out = SiLU( (agg + x*dis^2) @ W + b )  via V_WMMA_F32_16X16X4_F32
// 256 threads = 8 wave32; block does a 16x128 slab, each wave one 16x16 tile.
// W (64KB) + folded A tile (8.25KB) staged in LDS with stride-132 padding.
// ---------------------------------------------------------------------------
__global__ __launch_bounds__(256)
void gcn_gemm_wmma(const float* __restrict__ agg, const float* __restrict__ x,
                   const float* __restrict__ dis,
                   const float* __restrict__ W, const float* __restrict__ bias,
                   float* __restrict__ out) {
    __shared__ float Wl[CIN * LDS_STRIDE];   // 67,584 B
    __shared__ float Al[16 * LDS_STRIDE];    //  8,448 B

    int tid  = threadIdx.x;
    int row0 = blockIdx.x * 16;              // N = 100000 = 6250 * 16, exact

    // Stage W (row-major 128x128) into LDS
    for (int i = tid; i < CIN * COUT; i += 256) {
        int r = i >> 7, c = i & 127;
        Wl[r * LDS_STRIDE + c] = W[i];
    }
    // Stage A tile with self-loop term folded in: A = agg + x * dis^2
    for (int i = tid; i < 16 * CIN; i += 256) {
        int r = i >> 7, c = i & 127;
        int node = row0 + r;
        float dv = dis[node];
        size_t off = (size_t)node * CIN + c;
        Al[r * LDS_STRIDE + c] = agg[off] + x[off] * dv * dv;
    }
    __syncthreads();

    int wave = tid >> 5;           // 0..7  -> output cols [16*wave, 16*wave+16)
    int lane = tid & 31;
    int half = lane >> 4;          // 0: K pair {0,1}; 1: K pair {2,3}
    int ln   = lane & 15;
    int cb   = wave * 16;

    v8f acc = {};
    // K = 128 -> 32 WMMA steps of 16x16x4 fp32
    for (int k = 0; k < 32; ++k) {
        int k4 = k * 4 + half * 2;
        v2f a, bf;
        // A 16x4: lane ln = row, V0/V1 = two consecutive K in this half
        a.x = Al[ln * LDS_STRIDE + k4];
        a.y = Al[ln * LDS_STRIDE + k4 + 1];
        // B 4x16: lane ln = col, V0/V1 = two consecutive K rows in this half
        bf.x = Wl[k4 * LDS_STRIDE + cb + ln];
        bf.y = Wl[(k4 + 1) * LDS_STRIDE + cb + ln];
        acc = __builtin_amdgcn_wmma_f32_16x16x4_f32(
            /*neg_a=*/false, a, /*neg_b=*/false, bf,
            /*c_mod=*/(short)0, acc, /*reuse_a=*/false, /*reuse_b=*/false);
    }

    // Epilogue: C/D layout -> row = i + 8*half, col = cb + ln. Bias + SiLU.
    int col = cb + ln;
    float bv = bias[col];
    #pragma unroll
    for (int i = 0; i < 8; ++i) {
        int r = i + half * 8;
        float v = acc[i] + bv;
        v = v / (1.0f + __expf(-v));            // SiLU
        out[(size_t)(row0 + r) * COUT + col] = v;
    }
}

// ---------------------------------------------------------------------------
extern "C" void kernel_launch(void* const* d_in, const int* in_sizes, int n_in,
                              void* d_out, int out_size, void* d_ws, size_t ws_size,
                              hipStream_t stream) {
    const float* x  = (const float*)d_in[0];
    const int*   ei = (const int*)d_in[1];       // [2, E] int32
    const float* W  = (const float*)d_in[2];
    const float* b  = (const float*)d_in[3];
    float* out = (float*)d_out;

    const int N = in_sizes[0] / CIN;             // 100000
    const int E = in_sizes[1] / 2;               // 1600000
    const int* src = ei;
    const int* dst = ei + E;

    // Workspace: agg [N*128] fp32 (51.2 MB) + deg/dis [N] (0.4 MB)
    float* agg = (float*)d_ws;
    float* degdis = agg + (size_t)N * CIN;

    {   // init
        long long total = (long long)N * CIN;
        int blocks = (int)((total + 255) / 256);
        gcn_init<<<blocks, 256, 0, stream>>>(agg, (unsigned int*)degdis, N);
    }
    {   // degree
        int blocks = (E + 255) / 256;
        gcn_degree<<<blocks, 256, 0, stream>>>(dst, (unsigned int*)degdis, E);
    }
    {   // deg^{-1/2}
        int blocks = (N + 255) / 256;
        gcn_rsqrt<<<blocks, 256, 0, stream>>>(degdis, N);
    }
    {   // edge scatter: one wave32 per edge
        long long threads = (long long)E * 32;
        int blocks = (int)((threads + 255) / 256);
        gcn_scatter<<<blocks, 256, 0, stream>>>(x, src, dst, degdis, agg, E);
    }
    {   // WMMA GEMM + bias + SiLU
        int blocks = N / 16;                     // 6250
        gcn_gemm_wmma<<<blocks, 256, 0, stream>>>(agg, x, degdis, W, b, out);
    }
}